// ColorCorrectionLoss_24232205484622
// MI455X (gfx1250) — compile-verified
//
#include <hip/hip_runtime.h>

// ColorCorrectionLoss: mean |Lab8(pred) - Lab8(ref)| over (B=32,3,512,512) f32.
// Bandwidth-bound streaming kernel + deterministic 2-pass reduction.
// Wave reduction uses V_WMMA_F32_16X16X4_F32 (A = lane partials, B = ones).

#define HW_   262144   // 512*512
#define HW4_  65536    // HW/4 (float4 groups per plane)
#define NBLK_ 2048
#define NTHR_ 256

typedef __attribute__((ext_vector_type(2))) float v2f;
typedef __attribute__((ext_vector_type(8))) float v8f;

// ---- wave32 sum via WMMA f32 16x16x4 -------------------------------------
// A layout (16x4 f32, 2 VGPRs/lane): lanes 0-15 hold (M=lane,K=0),(M=lane,K=1);
// lanes 16-31 hold (M=lane-16,K=2),(M=lane-16,K=3).  With a1=0 and B=ones:
// D[m][n] = v_m + v_{m+16}.  D VGPR r: lanes<16 -> M=r, lanes>=16 -> M=r+8.
// Sum of 8 D VGPRs gives half-wave totals; one shfl_xor(16) completes it.
__device__ __forceinline__ float wave_sum32(float v) {
    v2f a; a[0] = v;    a[1] = 0.0f;
    v2f b; b[0] = 1.0f; b[1] = 1.0f;
    v8f c = {0.f, 0.f, 0.f, 0.f, 0.f, 0.f, 0.f, 0.f};
    v8f d = __builtin_amdgcn_wmma_f32_16x16x4_f32(false, a, false, b,
                                                  (short)0, c, false, false);
    float s = ((d[0] + d[1]) + (d[2] + d[3])) + ((d[4] + d[5]) + (d[6] + d[7]));
    s += __shfl_xor(s, 16, 32);
    return s;
}

// ---- Lab math -------------------------------------------------------------
#define T_LAB 0.008856f

__device__ __forceinline__ float lab_f(float t) {
    float ts = fmaxf(t, T_LAB);                       // safe input like reference
    float cb = __builtin_amdgcn_exp2f(0.33333333f * __builtin_amdgcn_logf(ts));
    return (t > T_LAB) ? cb : fmaf(7.787f, t, 0.13793103f);  // 16/116
}

__device__ __forceinline__ float clamp01(float v) {
    return __builtin_amdgcn_fmed3f(fmaf(v, 0.5f, 0.5f), 0.0f, 1.0f);
}

// fx, fy, fz and scaled L (offsets +128 cancel in |diff| and are dropped)
__device__ __forceinline__ void rgb_to_labf(float r, float g, float b,
                                            float& L, float& fx, float& fy, float& fz) {
    // X/XN and Z/ZN folded into the matrix (compile-time folded divisions)
    const float xr = 0.412453f / 0.950456f, xg = 0.35758f  / 0.950456f, xb = 0.180423f / 0.950456f;
    const float yr = 0.212671f,             yg = 0.71516f,              yb = 0.072169f;
    const float zr = 0.019334f / 1.088754f, zg = 0.119193f / 1.088754f, zb = 0.950227f / 1.088754f;
    float x = fmaf(xr, r, fmaf(xg, g, xb * b));
    float y = fmaf(yr, r, fmaf(yg, g, yb * b));
    float z = fmaf(zr, r, fmaf(zg, g, zb * b));
    fx = lab_f(x); fy = lab_f(y); fz = lab_f(z);
    float Lc = (y > T_LAB) ? fmaf(116.0f, fy, -16.0f) : (903.3f * y);
    L = Lc * 2.55f;                                   // *255/100
}

__device__ __forceinline__ float pixel_diff(float pr, float pg, float pb,
                                            float rr, float rg, float rb) {
    float Lp, fxp, fyp, fzp, Lr, fxr, fyr, fzr;
    rgb_to_labf(clamp01(pr), clamp01(pg), clamp01(pb), Lp, fxp, fyp, fzp);
    rgb_to_labf(clamp01(rr), clamp01(rg), clamp01(rb), Lr, fxr, fyr, fzr);
    float dL = fabsf(Lp - Lr);
    float da = 500.0f * fabsf((fxp - fyp) - (fxr - fyr));
    float db = 200.0f * fabsf((fyp - fzp) - (fyr - fzr));
    return dL + da + db;
}

// ---- kernel 1: per-block partial sums ------------------------------------
__global__ __launch_bounds__(NTHR_)
void lab_loss_partial(const float* __restrict__ pred, const float* __restrict__ ref,
                      float* __restrict__ bsum, int batch) {
    const unsigned total4 = (unsigned)batch * HW4_;
    const unsigned stride = gridDim.x * blockDim.x;
    float acc = 0.0f;
    for (unsigned i = blockIdx.x * blockDim.x + threadIdx.x; i < total4; i += stride) {
        unsigned b = i >> 16;                 // i / HW4_
        unsigned j = i & (HW4_ - 1);          // i % HW4_
        size_t base = (size_t)b * (3u * HW_) + ((size_t)j << 2);
        float4 pr = *reinterpret_cast<const float4*>(pred + base);
        float4 pg = *reinterpret_cast<const float4*>(pred + base + HW_);
        float4 pb = *reinterpret_cast<const float4*>(pred + base + 2u * HW_);
        float4 rr = *reinterpret_cast<const float4*>(ref  + base);
        float4 rg = *reinterpret_cast<const float4*>(ref  + base + HW_);
        float4 rb = *reinterpret_cast<const float4*>(ref  + base + 2u * HW_);
        acc += pixel_diff(pr.x, pg.x, pb.x, rr.x, rg.x, rb.x);
        acc += pixel_diff(pr.y, pg.y, pb.y, rr.y, rg.y, rb.y);
        acc += pixel_diff(pr.z, pg.z, pb.z, rr.z, rg.z, rb.z);
        acc += pixel_diff(pr.w, pg.w, pb.w, rr.w, rg.w, rb.w);
    }
    // wave32 reduction via WMMA, then cross-wave via LDS (fixed order)
    float wsum = wave_sum32(acc);
    __shared__ float lds[NTHR_ / 32];
    unsigned wid = threadIdx.x >> 5;
    if ((threadIdx.x & 31u) == 0u) lds[wid] = wsum;
    __syncthreads();
    if (threadIdx.x == 0) {
        float t = 0.0f;
        #pragma unroll
        for (int w = 0; w < NTHR_ / 32; ++w) t += lds[w];
        bsum[blockIdx.x] = t;
    }
}

// ---- kernel 2: deterministic final reduce + scale ------------------------
__global__ __launch_bounds__(NTHR_)
void lab_loss_final(const float* __restrict__ bsum, int n,
                    float* __restrict__ out, float scale) {
    float acc = 0.0f;
    for (int i = threadIdx.x; i < n; i += NTHR_) acc += bsum[i];
    float wsum = wave_sum32(acc);
    __shared__ float lds[NTHR_ / 32];
    unsigned wid = threadIdx.x >> 5;
    if ((threadIdx.x & 31u) == 0u) lds[wid] = wsum;
    __syncthreads();
    if (threadIdx.x == 0) {
        float t = 0.0f;
        #pragma unroll
        for (int w = 0; w < NTHR_ / 32; ++w) t += lds[w];
        out[0] = t * scale;
    }
}

extern "C" void kernel_launch(void* const* d_in, const int* in_sizes, int n_in,
                              void* d_out, int out_size, void* d_ws, size_t ws_size,
                              hipStream_t stream) {
    const float* pred = (const float*)d_in[0];
    const float* ref  = (const float*)d_in[1];
    int batch = in_sizes[0] / (3 * HW_);      // 32 for the reference shapes
    float* bsum = (float*)d_ws;               // NBLK_ floats = 8 KB scratch

    lab_loss_partial<<<NBLK_, NTHR_, 0, stream>>>(pred, ref, bsum, batch);

    float scale = 1.0f / ((float)batch * 3.0f * (float)HW_);  // mean over (B,3,H,W)
    lab_loss_final<<<1, NTHR_, 0, stream>>>(bsum, NBLK_, (float*)d_out, scale);
}